// Repro_39865886442252
// MI455X (gfx1250) — compile-verified
//
#include <hip/hip_runtime.h>

// Antialiased horizontal resize 456 -> 272 as a banded GEMM on the CDNA5
// matrix engine: out[1035 x 272] = x[1035 x 456] * W[456 x 272], band <= 4,
// one 16x16 output tile per wave32 via eight chained V_WMMA_F32_16X16X4_F32.
// Fully branchless inner loop: the band weight for (input col, output col) is
// evaluated analytically and gated with a single cndmask, so the only control
// flow is the scalar ragged-row check at the stores.

typedef __attribute__((ext_vector_type(2))) float v2f;
typedef __attribute__((ext_vector_type(8))) float v8f;

#define OUT_W 272
#define IN_W  456
#define ROWS  1035                      // C*H = 3*345
#define KS    5
#define SCALE_F     1.6764705882352942f // 456/272
#define INV_SCALE_F 0.5964912280701754f // 272/456

__global__ __launch_bounds__(32) void resize_wmma_f32(const float* __restrict__ x,
                                                      float* __restrict__ out) {
    const int lane = threadIdx.x;       // 0..31, one wave per block
    const int n    = lane & 15;         // output column within tile / A-matrix row
    const int hi   = lane >> 4;         // half-wave selector for K striping
    const int ow0  = blockIdx.x * 16;   // output-column tile base
    const int row0 = blockIdx.y * 16;   // (c,h)-row tile base

    // ---- per-output-column filter support (reference semantics: trunc casts)
    const int   ow = ow0 + n;
    const float fc = ((float)ow + 0.5f) * SCALE_F;
    int xmin = (int)(fc - SCALE_F + 0.5f); if (xmin < 0) xmin = 0;
    int xmax = (int)(fc + SCALE_F + 0.5f); if (xmax > IN_W) xmax = IN_W;
    int size = xmax - xmin;                if (size > KS) size = KS;

    // Normalization sum over masked taps j = 0..4 (branchless, reference order).
    float wsum = 0.0f;
#pragma unroll
    for (int j = 0; j < KS; ++j) {
        float t = ((float)(j + xmin) - fc + 0.5f) * INV_SCALE_F;
        float w = 1.0f - fminf(fabsf(t), 1.0f);
        wsum += (j < size) ? w : 0.0f;
    }
    const float inv_s = 1.0f / wsum;
    // Taps j >= 4 are zeroed post-normalization by the reference; size <= 4 in
    // practice, so the effective tap window is j in [0, smax).
    const unsigned smax = (unsigned)((size > 4) ? 4 : size);

    // ---- input-column window for this tile, clamped so the whole 32-wide K
    // window is in-bounds (all nonzero taps still land in k in [0,32)).
    const float fc0 = ((float)ow0 + 0.5f) * SCALE_F;
    int iw0 = (int)(fc0 - SCALE_F + 0.5f);
    if (iw0 < 0) iw0 = 0;
    if (iw0 > IN_W - 32) iw0 = IN_W - 32;
    const int shift = xmin - iw0;       // this column's band offset within window

    // ---- A operand pointer: lane holds x[row0 + n][iw0 + 2*hi + 4*kk + {0,1}]
    int r = row0 + n; if (r >= ROWS) r = ROWS - 1;   // clamp; masked at store
    const float* __restrict__ p = x + (size_t)r * IN_W + iw0 + 2 * hi;

    v8f acc = {};                       // C/D: 16x16 f32 tile in 8 VGPRs

#pragma unroll
    for (int kk = 0; kk < 8; ++kk) {    // K = 32 band, 8 steps of K = 4
        const int k0 = kk * 4 + hi * 2; // this lane's K pair: {k0, k0+1}

        // A: two consecutive input pixels (merges to global_load_b64 + imm off)
        v2f a;
        a.x = p[kk * 4];
        a.y = p[kk * 4 + 1];

        // B: analytic triangle weight, gated by tap-in-support (one cndmask each)
        const int   iw = iw0 + k0;
        const float t0 = ((float)iw - fc + 0.5f) * INV_SCALE_F;
        const float t1 = ((float)(iw + 1) - fc + 0.5f) * INV_SCALE_F;
        const float q0 = (1.0f - fminf(fabsf(t0), 1.0f)) * inv_s;
        const float q1 = (1.0f - fminf(fabsf(t1), 1.0f)) * inv_s;
        const int   j0 = k0 - shift;
        v2f b;
        b.x = ((unsigned)j0 < smax)       ? q0 : 0.0f;
        b.y = ((unsigned)(j0 + 1) < smax) ? q1 : 0.0f;

        // D = A(16x4) x B(4x16) + C ; EXEC is all-1s (full wave, no divergence)
        acc = __builtin_amdgcn_wmma_f32_16x16x4_f32(
            /*neg_a=*/false, a, /*neg_b=*/false, b,
            /*c_mod=*/(short)0, acc, /*reuse_a=*/false, /*reuse_b=*/false);
    }

    // ---- store D: lane L, VGPR v -> out[row0 + v + 8*hi][ow0 + n]
    float* __restrict__ o = out + (size_t)(row0 + 8 * hi) * OUT_W + ow0 + n;
    if (row0 + 16 <= ROWS) {            // scalar-uniform fast path (64/65 tiles)
#pragma unroll
        for (int v = 0; v < 8; ++v) {
            o[(size_t)v * OUT_W] = acc[v];
        }
    } else {                            // ragged last row-tile
#pragma unroll
        for (int v = 0; v < 8; ++v) {
            if (row0 + v + 8 * hi < ROWS) {
                o[(size_t)v * OUT_W] = acc[v];
            }
        }
    }
}

extern "C" void kernel_launch(void* const* d_in, const int* in_sizes, int n_in,
                              void* d_out, int out_size, void* d_ws, size_t ws_size,
                              hipStream_t stream) {
    (void)in_sizes; (void)n_in; (void)d_ws; (void)ws_size; (void)out_size;
    const float* x = (const float*)d_in[0];
    float* out     = (float*)d_out;

    dim3 grid((OUT_W + 15) / 16,        // 17 column tiles
              (ROWS + 15) / 16);        // 65 row tiles
    dim3 block(32);                     // exactly one wave32 -> EXEC all-1s for WMMA
    resize_wmma_f32<<<grid, block, 0, stream>>>(x, out);
}